// MonarchScoreCalculator_24558622998776
// MI455X (gfx1250) — compile-verified
//
#include <hip/hip_runtime.h>
#include <hip/hip_bf16.h>

// Problem constants (match reference)
#define Bb   4
#define Ss   1024
#define Hh   8
#define Dd   64
#define SN   32
#define NC   4
#define NBH  (Bb*Hh)        // 32
#define KDIM (NC*Dd)        // 256 folded reduction dim

typedef __attribute__((ext_vector_type(2))) float v2f;
typedef __attribute__((ext_vector_type(8))) float v8f;

// ---------------------------------------------------------------------------
// Kernel 1: Monarch transform.
//   y = S_p( S_32( BD(R) * S_32( BD(L) * S_32( S_p( x ) ) ) ) )
// with swap(x,q): out[o] = in[(o % q)*(n/q) + o/q].
// Each block: one (bh, comp, 4 d-rows), blockIdx.y: 0 -> Q (M1, softmax scale),
// 1 -> K (M2, scale 1). Output layout: [bh][c*64+d][s] (k-major rows, s contig).
// ---------------------------------------------------------------------------
#define DC 4
__global__ __launch_bounds__(256) void monarch_transform(
    const float* __restrict__ q,  const float* __restrict__ k,
    const float* __restrict__ M1L, const float* __restrict__ M1R,
    const float* __restrict__ M2L, const float* __restrict__ M2R,
    const float* __restrict__ wts, const int* __restrict__ periods,
    float* __restrict__ Qp, float* __restrict__ Kp)
{
    __shared__ float bufA[DC][1024];
    __shared__ float bufB[DC][1024];

    const int bx     = blockIdx.x;
    const int dchunk = bx & 15;          // 64/DC = 16 chunks
    const int comp   = (bx >> 4) & 3;
    const int bh     = bx >> 6;          // 0..31
    const int b      = bh >> 3;
    const int h      = bh & 7;
    const int d0     = dchunk * DC;
    const int tid    = threadIdx.x;
    const bool isQ   = (blockIdx.y == 0);

    const float* __restrict__ x   = isQ ? q   : k;
    const float* __restrict__ WL  = isQ ? M1L : M2L;
    const float* __restrict__ WR  = isQ ? M1R : M2R;
    float*       __restrict__ dst = isQ ? Qp  : Kp;

    // softmax(weights[b,:]) -> scale for this comp (Q side only)
    float scale = 1.0f;
    if (isQ) {
        float w0 = wts[b*NC+0], w1 = wts[b*NC+1], w2 = wts[b*NC+2], w3 = wts[b*NC+3];
        float mx = fmaxf(fmaxf(w0, w1), fmaxf(w2, w3));
        float e0 = expf(w0-mx), e1 = expf(w1-mx), e2 = expf(w2-mx), e3 = expf(w3-mx);
        float s  = e0 + e1 + e2 + e3;
        float ec = (comp == 0) ? e0 : (comp == 1) ? e1 : (comp == 2) ? e2 : e3;
        scale = ec / s;
    }

    const int p  = periods[comp];                 // {1,2,4,8}, powers of two
    const int lp = 31 - __builtin_clz((unsigned)p);
    const int f  = 1024 >> lp;                    // n / p

    // Stage A: bufA[dl][j] = x[ srcp(src32(j)) ]   (gather from global)
    for (int e = tid; e < DC*1024; e += 256) {
        const int dl = e >> 10;
        const int j  = e & 1023;
        const int t  = ((j & 31) << 5) | (j >> 5);           // src32
        const int sj = (t & (p-1)) * f + (t >> lp);          // srcp
        bufA[dl][j] = x[((size_t)(b*Ss + sj)*Hh + h)*Dd + d0 + dl];
    }
    __syncthreads();

    // Stage B: bufB[dl][blk*32+n] = sum_m L[comp][blk][n][m] * bufA[dl][blk*32+m]
    for (int pr = tid; pr < 1024; pr += 256) {
        const int blk = pr >> 5;
        const int n   = pr & 31;
        const float* __restrict__ wrow = WL + (((size_t)comp*SN + blk)*SN + n)*SN;
        float acc[DC] = {0.f, 0.f, 0.f, 0.f};
        #pragma unroll 8
        for (int m = 0; m < SN; ++m) {
            const float wv = wrow[m];
            #pragma unroll
            for (int dl = 0; dl < DC; ++dl) acc[dl] += wv * bufA[dl][blk*32 + m];
        }
        #pragma unroll
        for (int dl = 0; dl < DC; ++dl) bufB[dl][blk*32 + n] = acc[dl];
    }
    __syncthreads();

    // Stage C+D fused: xd[blk*32+n] = sum_m R[blk][n][m] * bufB[dl][m*32+blk]
    for (int pr = tid; pr < 1024; pr += 256) {
        const int blk = pr >> 5;
        const int n   = pr & 31;
        const float* __restrict__ wrow = WR + (((size_t)comp*SN + blk)*SN + n)*SN;
        float acc[DC] = {0.f, 0.f, 0.f, 0.f};
        #pragma unroll 8
        for (int m = 0; m < SN; ++m) {
            const float wv = wrow[m];
            #pragma unroll
            for (int dl = 0; dl < DC; ++dl) acc[dl] += wv * bufB[dl][m*32 + blk];
        }
        #pragma unroll
        for (int dl = 0; dl < DC; ++dl) bufA[dl][blk*32 + n] = acc[dl];
    }
    __syncthreads();

    // Stage E: y[j] = xd[ src32(srcp(j)) ] * scale ; coalesced store over j
    for (int e = tid; e < DC*1024; e += 256) {
        const int dl = e >> 10;
        const int j  = e & 1023;
        const int u  = (j & (p-1)) * f + (j >> lp);          // srcp
        const int ix = ((u & 31) << 5) | (u >> 5);           // src32
        dst[((size_t)(bh*NC + comp)*Dd + d0 + dl)*1024 + j] = bufA[dl][ix] * scale;
    }
}

// ---------------------------------------------------------------------------
// Kernel 2: batched GEMM  out[bh][s][l] = sum_k Q'[bh][k][s] * K'[bh][k][l]
// 128x128 block tile, KC=32 LDS staging, 8 waves of 32x64, f32 WMMA 16x16x4.
// LDS tiles are K-pair interleaved: element (k, m) lives at
//   T[k>>1][m*2 + (k&1)]
// so each WMMA fragment (.x = K even, .y = K odd, same column) is one aligned
// conflict-free ds_load_b64 (row pitch 288: 288 mod 64 = 32 -> the two
// half-wave rows use complementary bank sets).
// ---------------------------------------------------------------------------
#define TM  128
#define TN  128
#define KC  32
#define KPH (KC/2)          // 16 k-pair rows
#define LDP (2*128 + 32)    // 288 floats per k-pair row

__global__ __launch_bounds__(256) void score_gemm(
    const float* __restrict__ Qp, const float* __restrict__ Kp,
    float* __restrict__ out)
{
    __shared__ __align__(16) float As[KPH][LDP];
    __shared__ __align__(16) float Bs[KPH][LDP];

    const int bh  = blockIdx.z;
    const int m0  = blockIdx.y * TM;
    const int n0  = blockIdx.x * TN;
    const int tid = threadIdx.x;
    const int lane = tid & 31;
    const int wave = tid >> 5;
    const int wm = wave & 3;             // 4 waves along M (4*32 = 128)
    const int wn = wave >> 2;            // 2 waves along N (2*64 = 128)
    const int lm = lane & 15;
    const int lkp = lane >> 4;           // k-pair sub-offset: 0 / 1

    const float* __restrict__ Ag = Qp + (size_t)bh * KDIM * 1024;
    const float* __restrict__ Bg = Kp + (size_t)bh * KDIM * 1024;

    v8f acc[2][4];
    #pragma unroll
    for (int i = 0; i < 2; ++i)
        #pragma unroll
        for (int j = 0; j < 4; ++j) acc[i][j] = (v8f){0.f,0.f,0.f,0.f,0.f,0.f,0.f,0.f};

    for (int kc = 0; kc < KDIM; kc += KC) {
        // Stage KCx128 panels of A and B into LDS, interleaving K pairs.
        // Each thread: two row-float4 loads per matrix, two interleaved
        // float4 LDS stores per matrix. Fully coalesced global traffic.
        #pragma unroll
        for (int i = 0; i < 2; ++i) {
            const int idx = tid + i * 256;        // 0..511
            const int kp  = idx >> 5;             // 0..15
            const int c4  = (idx & 31) * 4;       // 0..124
            const size_t r0 = (size_t)(kc + 2*kp    ) * 1024;
            const size_t r1 = (size_t)(kc + 2*kp + 1) * 1024;

            const float4 a0 = *(const float4*)(Ag + r0 + m0 + c4);
            const float4 a1 = *(const float4*)(Ag + r1 + m0 + c4);
            *(float4*)&As[kp][c4*2    ] = make_float4(a0.x, a1.x, a0.y, a1.y);
            *(float4*)&As[kp][c4*2 + 4] = make_float4(a0.z, a1.z, a0.w, a1.w);

            const float4 b0 = *(const float4*)(Bg + r0 + n0 + c4);
            const float4 b1 = *(const float4*)(Bg + r1 + n0 + c4);
            *(float4*)&Bs[kp][c4*2    ] = make_float4(b0.x, b1.x, b0.y, b1.y);
            *(float4*)&Bs[kp][c4*2 + 4] = make_float4(b0.z, b1.z, b0.w, b1.w);
        }
        __syncthreads();

        #pragma unroll
        for (int kk = 0; kk < KC; kk += 4) {
            const int kp = (kk >> 1) + lkp;       // (kk + 2*(lane>=16)) / 2
            v2f af[2], bf[4];
            #pragma unroll
            for (int i = 0; i < 2; ++i)
                af[i] = *(const v2f*)&As[kp][2*(wm*32 + i*16 + lm)];
            #pragma unroll
            for (int j = 0; j < 4; ++j)
                bf[j] = *(const v2f*)&Bs[kp][2*(wn*64 + j*16 + lm)];
            #pragma unroll
            for (int i = 0; i < 2; ++i)
                #pragma unroll
                for (int j = 0; j < 4; ++j)
                    acc[i][j] = __builtin_amdgcn_wmma_f32_16x16x4_f32(
                        /*neg_a=*/false, af[i], /*neg_b=*/false, bf[j],
                        /*c_mod=*/(short)0, acc[i][j],
                        /*reuse_a=*/false, /*reuse_b=*/false);
        }
        __syncthreads();
    }

    // Store: D element r -> (M = r + 8*(lane>=16), N = lane&15). Non-temporal:
    // 128 MB of output must not evict the L2-resident Q'/K' workspace.
    float* __restrict__ ob = out + (size_t)bh * 1024 * 1024;
    #pragma unroll
    for (int i = 0; i < 2; ++i) {
        #pragma unroll
        for (int j = 0; j < 4; ++j) {
            const int l = n0 + wn*64 + j*16 + lm;
            #pragma unroll
            for (int r = 0; r < 8; ++r) {
                const int s = m0 + wm*32 + i*16 + r + 8*(lane >> 4);
                __builtin_nontemporal_store(acc[i][j][r], ob + (size_t)s*1024 + l);
            }
        }
    }
}

// ---------------------------------------------------------------------------
extern "C" void kernel_launch(void* const* d_in, const int* in_sizes, int n_in,
                              void* d_out, int out_size, void* d_ws, size_t ws_size,
                              hipStream_t stream) {
    const float* q   = (const float*)d_in[0];
    const float* k   = (const float*)d_in[1];
    const float* M1L = (const float*)d_in[2];
    const float* M1R = (const float*)d_in[3];
    const float* M2L = (const float*)d_in[4];
    const float* M2R = (const float*)d_in[5];
    const float* w   = (const float*)d_in[6];
    const int*   per = (const int*)d_in[7];
    float* out = (float*)d_out;

    // Workspace: Q' then K', each [32][256][1024] f32 = 32 MB (64 MB total,
    // L2-resident on MI455X's 192 MB L2).
    float* Qp = (float*)d_ws;
    float* Kp = Qp + (size_t)NBH * KDIM * 1024;

    monarch_transform<<<dim3(NBH*NC*(Dd/DC), 2), 256, 0, stream>>>(
        q, k, M1L, M1R, M2L, M2R, w, per, Qp, Kp);
    score_gemm<<<dim3(1024/TN, 1024/TM, NBH), 256, 0, stream>>>(
        Qp, Kp, out);
}